// EnhancedProbabilisticCausalPFNO_25718264169337
// MI455X (gfx1250) — compile-verified
//
#include <hip/hip_runtime.h>
#include <math.h>

typedef float v2f __attribute__((ext_vector_type(2)));
typedef float v8f __attribute__((ext_vector_type(8)));

#define EPSV 1e-5f

__device__ __forceinline__ v8f wmma_f32(v2f a, v2f b, v8f c) {
  // D = A(16x4 f32) * B(4x16 f32) + C  -> v_wmma_f32_16x16x4_f32
  return __builtin_amdgcn_wmma_f32_16x16x4_f32(false, a, false, b, (short)0, c, false, false);
}

__device__ __forceinline__ float gelu_f(float x) {
  return 0.5f * x * (1.0f + erff(x * 0.70710678118654752440f));
}

// ---------------------------------------------------------------------------
// Generic batched GEMM: C[z] = op(A[z]) * op(B[z]) with epilogue.
// One wave per 16x16 tile. K must be a multiple of 8 (two-accumulator loop).
// Out-of-range rows/cols are CLAMPED for loads (value-safe: A row r only feeds
// D row r, B col c only feeds D col c) and guarded at store time, so the K-loop
// has no EXEC-mask churn.
// ---------------------------------------------------------------------------
template <int TA, int TB>
__global__ void __launch_bounds__(32) gemm16(
    const float* __restrict__ A, long long sA, int lda,
    const float* __restrict__ B, long long sB, int ldb,
    float* __restrict__ C, long long sC, int ldc,
    int M, int N, int K,
    const float* __restrict__ biasM,
    const float* __restrict__ biasN,
    const float* __restrict__ scale, const float* __restrict__ shift, float scaleMul,
    const float* __restrict__ resid, long long sR, int ldr,
    int accumIn, int act)
{
  int lane = threadIdx.x & 31;
  int half = lane >> 4;
  int l16  = lane & 15;
  int n0 = blockIdx.x * 16;
  int m0 = blockIdx.y * 16;
  int z  = blockIdx.z;
  const float* Ap = A + (size_t)z * (size_t)sA;
  const float* Bp = B + (size_t)z * (size_t)sB;
  float*       Cp = C + (size_t)z * (size_t)sC;
  int arowS = m0 + l16;
  int bcolS = n0 + l16;
  int arow = arowS < M ? arowS : (M - 1);
  int bcol = bcolS < N ? bcolS : (N - 1);

  const float* ap;
  const float* bp;
  size_t astep, bstep;
  int aoff2, boff2;
  if (TA) { ap = Ap + (size_t)(half * 2) * lda + arow; astep = (size_t)lda * 4; aoff2 = lda; }
  else    { ap = Ap + (size_t)arow * lda + half * 2;   astep = 4;               aoff2 = 1;   }
  if (TB) { bp = Bp + (size_t)bcol * ldb + half * 2;   bstep = 4;               boff2 = 1;   }
  else    { bp = Bp + (size_t)(half * 2) * ldb + bcol; bstep = (size_t)ldb * 4; boff2 = ldb; }

  v8f acc0 = {0.f,0.f,0.f,0.f,0.f,0.f,0.f,0.f};
  v8f acc1 = {0.f,0.f,0.f,0.f,0.f,0.f,0.f,0.f};
  int nsteps = K >> 2;   // always even for our shapes
  for (int s = 0; s < nsteps; s += 2) {
    v2f a0 = {ap[0], ap[aoff2]};
    v2f b0 = {bp[0], bp[boff2]};
    ap += astep; bp += bstep;
    v2f a1 = {ap[0], ap[aoff2]};
    v2f b1 = {bp[0], bp[boff2]};
    ap += astep; bp += bstep;
    acc0 = wmma_f32(a0, b0, acc0);
    acc1 = wmma_f32(a1, b1, acc1);
  }
  v8f acc = acc0 + acc1;

  if (bcolS < N) {
    #pragma unroll
    for (int r = 0; r < 8; ++r) {
      int m = m0 + r + half * 8;
      if (m < M) {
        float v = acc[r];
        if (accumIn) v += Cp[(size_t)m*ldc + bcolS];
        if (biasM)   v += biasM[m];
        if (biasN)   v += biasN[bcolS];
        if (scale)   v = v * scale[m] * scaleMul + shift[m];
        if (resid)   v += resid[(size_t)z*(size_t)sR + (size_t)m*ldr + bcolS];
        if (act)     v = gelu_f(v);
        Cp[(size_t)m*ldc + bcolS] = v;
      }
    }
  }
}

// ---------------------------------------------------------------------------
// lift+pre folded to per-channel affine
// ---------------------------------------------------------------------------
__global__ void prep_lift(const float* __restrict__ lift_w, const float* __restrict__ lift_b,
                          const float* __restrict__ pre_w,  const float* __restrict__ pre_b,
                          float* __restrict__ wx, float* __restrict__ wb)
{
  int c = threadIdx.x;
  if (c < 64) {
    float sx = 0.f, sb = 0.f;
    for (int l = 0; l < 64; ++l) { float w = pre_w[c*64 + l]; sx += w * lift_w[l]; sb += w * lift_b[l]; }
    wx[c] = sx; wb[c] = sb + pre_b[c];
  }
}

__global__ void lift_apply(const float* __restrict__ x, const float* __restrict__ wx,
                           const float* __restrict__ wb, float* __restrict__ H)
{
  size_t i = (size_t)blockIdx.x * 256 + threadIdx.x;  // B*C*2048 = 524288
  if (i < 524288) {
    int p  = (int)(i & 2047);
    int bc = (int)(i >> 11);
    int c  = bc & 63;
    int b  = bc >> 6;
    H[i] = wx[c] * x[(size_t)b*2048 + p] + wb[c];
  }
}

// ---------------------------------------------------------------------------
// DFT bases over T=128 (forward rDFT: 128x64 [cos | -sin], ky<32)
// (inverse rDFT: 64x128, includes 1/T and Hermitian doubling; Im(ky=0) dropped)
// ---------------------------------------------------------------------------
__global__ void build_basis(float* __restrict__ bf, float* __restrict__ bi)
{
  int t = blockIdx.x;   // 0..127
  int j = threadIdx.x;  // 0..63
  const float TWO_PI = 6.28318530717958647692f;
  int ky = j & 31;
  bool isim = j >= 32;
  float ang = TWO_PI * (float)ky * (float)t / 128.0f;
  bf[t*64 + j] = isim ? -sinf(ang) : cosf(ang);
  float v;
  if (!isim) v = (ky == 0 ? 1.0f : 2.0f * cosf(ang)) * (1.0f/128.0f);
  else       v = (ky == 0 ? 0.0f : -2.0f * sinf(ang)) * (1.0f/128.0f);
  bi[j*128 + t] = v;
}

// ---------------------------------------------------------------------------
// 16-point complex DFT over N: out[kx][ky] = scale * sum_n in[n][ky] e^{sign*2pi i kx n/16}
// Layout per (b,c): [16][64] with re at ky, im at 32+ky. One block per (b,c).
// ---------------------------------------------------------------------------
__global__ void fft16n(const float* __restrict__ in, float* __restrict__ out, float sign, float scale)
{
  __shared__ float s[1024];
  const float* src = in  + (size_t)blockIdx.x * 1024;
  float*       dst = out + (size_t)blockIdx.x * 1024;
  for (int i = threadIdx.x; i < 1024; i += blockDim.x) s[i] = src[i];
  __syncthreads();
  int t = threadIdx.x;
  if (t < 512) {
    int kx = t >> 5, ky = t & 31;
    float re = 0.f, im = 0.f;
    for (int n = 0; n < 16; ++n) {
      float ang = 6.28318530717958647692f * (float)kx * (float)n / 16.0f;
      float sn, cs;
      __sincosf(ang, &sn, &cs);
      sn *= sign;
      float xr = s[n*64 + ky], xi = s[n*64 + 32 + ky];
      re += xr*cs - xi*sn;
      im += xr*sn + xi*cs;
    }
    dst[kx*64 + ky]      = re * scale;
    dst[kx*64 + 32 + ky] = im * scale;
  }
}

// ---------------------------------------------------------------------------
// Per-mode complex channel mixing: of[b,o,kx,ky] = sum_i xf[b,i,kx,ky]*w[i,o,kx,ky]
// ---------------------------------------------------------------------------
__global__ void modemix(const float* __restrict__ xf,
                        const float* __restrict__ wr1, const float* __restrict__ wi1,
                        const float* __restrict__ wr2, const float* __restrict__ wi2,
                        int l, float* __restrict__ of)
{
  int mode = blockIdx.x;           // 0..511
  int kx = mode >> 5, ky = mode & 31;
  __shared__ float xr[4][64], xi[4][64];
  for (int i = threadIdx.x; i < 256; i += blockDim.x) {
    int b = i >> 6, c = i & 63;
    size_t base = ((size_t)(b*64 + c)*16 + kx)*64 + ky;
    xr[b][c] = xf[base];
    xi[b][c] = xf[base + 32];
  }
  __syncthreads();
  int t = threadIdx.x;
  int b = t >> 6, o = t & 63;
  const float* WR; const float* WI; size_t wbase;
  if (kx < 8) { WR = wr1; WI = wi1; wbase = (size_t)l*64*64*256 + (size_t)kx*32 + ky; }
  else        { WR = wr2; WI = wi2; wbase = (size_t)l*64*64*256 + (size_t)(kx-8)*32 + ky; }
  float ar = 0.f, ai = 0.f;
  for (int i = 0; i < 64; ++i) {
    size_t wix = wbase + ((size_t)i*64 + o)*256;
    float wr = WR[wix], wim = WI[wix];
    float cr = xr[b][i], ci = xi[b][i];
    ar += cr*wr - ci*wim;
    ai += cr*wim + ci*wr;
  }
  size_t obase = ((size_t)(b*64 + o)*16 + kx)*64 + ky;
  of[obase]      = ar;
  of[obase + 32] = ai;
}

// ---------------------------------------------------------------------------
// Conv1d(K=10,'VALID') + BN(eval) + GELU via implicit im2col WMMA GEMM.
// Output ns[b][n][tau][o]  (tau<119). Incremental (channel, tap) counters
// replace per-iteration integer division.
// ---------------------------------------------------------------------------
__global__ void __launch_bounds__(32) conv_loc(const float* __restrict__ H, const float* __restrict__ W,
                         const float* __restrict__ cb, const float* __restrict__ bng,
                         const float* __restrict__ bnb, float* __restrict__ ns)
{
  int lane = threadIdx.x & 31;
  int half = lane >> 4, l16 = lane & 15;
  int n0 = blockIdx.x * 16;   // tau tile (8 tiles -> 0..127)
  int m0 = blockIdx.y * 16;   // o tile (4 tiles)
  int bn = blockIdx.z;        // 0..63
  int b = bn >> 4, n = bn & 15;
  const float* src = H + (size_t)b*64*2048 + (size_t)n*128;  // row c at +c*2048
  int orow = m0 + l16;
  int col  = n0 + l16;
  int colc = col > 118 ? 118 : col;
  const float* wp = W + (size_t)orow*640 + half*2;
  const float* s0 = src + colc;
  int ka = half * 2;
  int c0 = 0, r0 = ka;        // ka in {0,2} -> tap index
  int c1 = 0, r1 = ka + 1;
  v8f acc = {0.f,0.f,0.f,0.f,0.f,0.f,0.f,0.f};
  for (int s = 0; s < 160; ++s) {
    v2f av = {wp[0], wp[1]};
    v2f bv = {s0[(size_t)c0*2048 + r0], s0[(size_t)c1*2048 + r1]};
    acc = wmma_f32(av, bv, acc);
    wp += 4;
    r0 += 4; if (r0 >= 10) { r0 -= 10; ++c0; }
    r1 += 4; if (r1 >= 10) { r1 -= 10; ++c1; }
  }
  if (col < 119) {
    float sm = rsqrtf(1.0f + EPSV);
    #pragma unroll
    for (int r = 0; r < 8; ++r) {
      int o = m0 + r + half * 8;
      float v = acc[r] + cb[o];
      v = v * bng[o] * sm + bnb[o];
      v = gelu_f(v);
      ns[((size_t)bn*119 + col)*64 + o] = v;
    }
  }
}

// ---------------------------------------------------------------------------
// cl: transpose logits, LayerNorm over last dim, scale, mask diag + causal bias.
// Writes clbuf (for attention) and cm/cl sections of d_out.
// ---------------------------------------------------------------------------
__global__ void cl_kernel(const float* __restrict__ logits, const float* __restrict__ ln_g,
                          const float* __restrict__ ln_b, const float* __restrict__ logit_scale,
                          const float* __restrict__ causal_bias,
                          float* __restrict__ clbuf, float* __restrict__ out)
{
  int b = blockIdx.x;
  int t = threadIdx.x;           // 256
  int r = t >> 4, i = t & 15;
  const float* L = logits + b*256;   // L[i][j]; cl[b,r,i] = L[i][r]
  float mu = 0.f;
  for (int ii = 0; ii < 16; ++ii) mu += L[ii*16 + r];
  mu *= (1.0f/16.0f);
  float var = 0.f;
  for (int ii = 0; ii < 16; ++ii) { float d = L[ii*16 + r] - mu; var += d*d; }
  var *= (1.0f/16.0f);
  float xv = L[i*16 + r];
  float v = (xv - mu) * rsqrtf(var + EPSV) * ln_g[i] + ln_b[i];
  v *= logit_scale[0];
  float fin = (r == i) ? 0.f : (v + causal_bias[r*16 + i]);
  clbuf[b*256 + r*16 + i] = fin;
  out[16384 + 1024 + b*256 + r*16 + i] = fin;                               // cl
  out[16384 +        b*256 + r*16 + i] = (r == i) ? 0.f : 1.0f/(1.0f + __expf(-fin)); // cm
}

// ---------------------------------------------------------------------------
// Flash attention over NT=2048 tokens, Dh=16, per-(b,h,q-tile) wave.
// bias cl[b, qN, kN] is constant per 16x16 tile (128%16==0).
// ---------------------------------------------------------------------------
__global__ void __launch_bounds__(32) attn_flash(const float* __restrict__ qkv,
                                                 const float* __restrict__ clbuf,
                                                 float* __restrict__ attnOut)
{
  __shared__ float P[256];
  int lane = threadIdx.x;
  int half = lane >> 4, l16 = lane & 15;
  int q0 = blockIdx.x * 16;
  int h  = blockIdx.y;
  int b  = blockIdx.z;
  const float* base = qkv + (size_t)b * 2048 * 192;
  int qrow = q0 + l16;
  v2f qa[4];
  {
    const float* qp = base + (size_t)qrow*192 + h*16;
    #pragma unroll
    for (int j = 0; j < 4; ++j) {
      int d = j*4 + half*2;
      v2f tmp = {qp[d], qp[d+1]};
      qa[j] = tmp;
    }
  }
  int qn = q0 >> 7;
  const float* clb = clbuf + b*256 + qn*16;
  float mrow[8], lrow[8];
  v8f O = {0.f,0.f,0.f,0.f,0.f,0.f,0.f,0.f};
  #pragma unroll
  for (int r = 0; r < 8; ++r) { mrow[r] = -1e30f; lrow[r] = 0.f; }

  for (int k0 = 0; k0 < 2048; k0 += 16) {
    int kcol = k0 + l16;
    const float* kp = base + (size_t)kcol*192 + 64 + h*16;
    v8f S = {0.f,0.f,0.f,0.f,0.f,0.f,0.f,0.f};
    #pragma unroll
    for (int j = 0; j < 4; ++j) {
      int d = j*4 + half*2;
      v2f bv = {kp[d], kp[d+1]};
      S = wmma_f32(qa[j], bv, S);
    }
    float bias = clb[k0 >> 7];
    float pv[8], alpha[8];
    #pragma unroll
    for (int r = 0; r < 8; ++r) {
      float s = S[r] * 0.25f + bias;          // Dh^-0.5 = 0.25
      float rm = s;
      rm = fmaxf(rm, __shfl_xor(rm, 1));
      rm = fmaxf(rm, __shfl_xor(rm, 2));
      rm = fmaxf(rm, __shfl_xor(rm, 4));
      rm = fmaxf(rm, __shfl_xor(rm, 8));
      float nm = fmaxf(mrow[r], rm);
      float al = __expf(mrow[r] - nm);
      float p  = __expf(s - nm);
      float rs = p;
      rs += __shfl_xor(rs, 1);
      rs += __shfl_xor(rs, 2);
      rs += __shfl_xor(rs, 4);
      rs += __shfl_xor(rs, 8);
      lrow[r] = lrow[r] * al + rs;
      mrow[r] = nm;
      alpha[r] = al;
      pv[r] = p;
    }
    #pragma unroll
    for (int r = 0; r < 8; ++r) O[r] *= alpha[r];
    #pragma unroll
    for (int r = 0; r < 8; ++r) P[(r + half*8)*16 + l16] = pv[r];
    __syncthreads();
    asm volatile("s_wait_dscnt 0" ::: "memory");
    const float* vp = base + (size_t)k0*192 + 128 + h*16 + l16;  // col=d=l16
    #pragma unroll
    for (int j = 0; j < 4; ++j) {
      int kk = j*4 + half*2;
      v2f av = {P[l16*16 + kk], P[l16*16 + kk + 1]};
      v2f bv = {vp[(size_t)kk*192], vp[(size_t)(kk+1)*192]};
      O = wmma_f32(av, bv, O);
    }
    __syncthreads();
  }
  #pragma unroll
  for (int r = 0; r < 8; ++r) {
    int row = q0 + r + half*8;
    attnOut[((size_t)b*2048 + row)*64 + h*16 + l16] = O[r] / lrow[r];
  }
}

__global__ void split_out(const float* __restrict__ zz3, float* __restrict__ out)
{
  int i = blockIdx.x * 256 + threadIdx.x;   // 16384
  if (i < 16384) {
    int p   = i & 2047;
    int rb  = i >> 11;
    int row = rb & 1;
    int b   = rb >> 1;
    out[row*8192 + b*2048 + p] = zz3[(size_t)b*4096 + row*2048 + p];
  }
}

// ---------------------------------------------------------------------------
extern "C" void kernel_launch(void* const* d_in, const int* in_sizes, int n_in,
                              void* d_out, int out_size, void* d_ws, size_t ws_size,
                              hipStream_t stream)
{
  (void)in_sizes; (void)n_in; (void)out_size; (void)ws_size;
  const float* x       = (const float*)d_in[0];
  const float* lift_w  = (const float*)d_in[1];
  const float* lift_b  = (const float*)d_in[2];
  const float* pre_w   = (const float*)d_in[3];
  const float* pre_b   = (const float*)d_in[4];
  const float* swr1    = (const float*)d_in[5];
  const float* swi1    = (const float*)d_in[6];
  const float* swr2    = (const float*)d_in[7];
  const float* swi2    = (const float*)d_in[8];
  const float* skip_w  = (const float*)d_in[9];
  const float* skip_b  = (const float*)d_in[10];
  const float* mlp_w1  = (const float*)d_in[11];
  const float* mlp_b1  = (const float*)d_in[12];
  const float* mlp_w2  = (const float*)d_in[13];
  const float* mlp_b2  = (const float*)d_in[14];
  const float* conv_w  = (const float*)d_in[15];
  const float* conv_b  = (const float*)d_in[16];
  const float* bn_g    = (const float*)d_in[17];
  const float* bn_b    = (const float*)d_in[18];
  const float* wq      = (const float*)d_in[19];
  const float* bq      = (const float*)d_in[20];
  const float* wk      = (const float*)d_in[21];
  const float* bk      = (const float*)d_in[22];
  const float* ln_g    = (const float*)d_in[23];
  const float* ln_b    = (const float*)d_in[24];
  const float* lsc     = (const float*)d_in[25];
  const float* cbias   = (const float*)d_in[26];
  const float* in_w    = (const float*)d_in[27];
  const float* in_b    = (const float*)d_in[28];
  const float* out_w   = (const float*)d_in[29];
  const float* out_b   = (const float*)d_in[30];
  const float* h1_w    = (const float*)d_in[31];
  const float* h1_b    = (const float*)d_in[32];
  const float* hbn1_g  = (const float*)d_in[33];
  const float* hbn1_b  = (const float*)d_in[34];
  const float* h2_w    = (const float*)d_in[35];
  const float* h2_b    = (const float*)d_in[36];
  const float* hbn2_g  = (const float*)d_in[37];
  const float* hbn2_b  = (const float*)d_in[38];
  const float* h3_w    = (const float*)d_in[39];
  const float* h3_b    = (const float*)d_in[40];
  float* out = (float*)d_out;

  float* w = (float*)d_ws;
  size_t off = 0;
  auto alloc = [&](size_t n) { float* p = w + off; off += n; return p; };
  float* WX   = alloc(64);
  float* WB   = alloc(64);
  float* BF   = alloc(128*64);
  float* BI   = alloc(64*128);
  float* H    = alloc(524288);    // [B,C,N,T]
  float* H2   = alloc(524288);
  float* SPEC = alloc(524288);
  float* MB   = alloc(524288);    // mlp intermediate, later attn-out projection
  float* ARI  = alloc(262144);    // [B*C, 16, 64] (re|im)
  float* XF   = alloc(262144);
  float* OF   = alloc(262144);
  float* FRI  = alloc(262144);
  float* NS   = alloc(487424);    // [B*N, 119, 64]
  float* QB   = alloc(487424);
  float* KB   = alloc(487424);
  float* LG   = alloc(1024);
  float* CLW  = alloc(1024);
  float* QKV  = alloc(1572864);   // [B, 2048, 192]
  float* AT   = alloc(524288);    // [B*2048, 64]
  float* Z1   = alloc(1048576);   // [B, 128, 2048]
  float* Z2   = alloc(524288);    // [B, 64, 2048]
  float* Z3   = alloc(16384);     // [B, 2, 2048]

  const float bnmul = 1.0f / sqrtf(1.0f + EPSV);
  const long long Hb = 131072;    // 64*2048 per-batch stride of [B,C,NT]

  auto gemm = [&](const float* A, long long sA, int lda, int tA,
                  const float* B, long long sB, int ldb, int tB,
                  float* C, long long sC, int ldc,
                  int M, int N, int K, int batch,
                  const float* biasM, const float* biasN,
                  const float* scale, const float* shift, float scaleMul,
                  const float* resid, long long sR, int ldr,
                  int accumIn, int act) {
    dim3 g((N + 15) / 16, (M + 15) / 16, batch);
    if (!tA && !tB)
      gemm16<0,0><<<g, 32, 0, stream>>>(A, sA, lda, B, sB, ldb, C, sC, ldc, M, N, K,
                                        biasM, biasN, scale, shift, scaleMul, resid, sR, ldr, accumIn, act);
    else if (!tA && tB)
      gemm16<0,1><<<g, 32, 0, stream>>>(A, sA, lda, B, sB, ldb, C, sC, ldc, M, N, K,
                                        biasM, biasN, scale, shift, scaleMul, resid, sR, ldr, accumIn, act);
    else if (tA && !tB)
      gemm16<1,0><<<g, 32, 0, stream>>>(A, sA, lda, B, sB, ldb, C, sC, ldc, M, N, K,
                                        biasM, biasN, scale, shift, scaleMul, resid, sR, ldr, accumIn, act);
    else
      gemm16<1,1><<<g, 32, 0, stream>>>(A, sA, lda, B, sB, ldb, C, sC, ldc, M, N, K,
                                        biasM, biasN, scale, shift, scaleMul, resid, sR, ldr, accumIn, act);
  };

  prep_lift<<<1, 64, 0, stream>>>(lift_w, lift_b, pre_w, pre_b, WX, WB);
  build_basis<<<128, 64, 0, stream>>>(BF, BI);
  lift_apply<<<2048, 256, 0, stream>>>(x, WX, WB, H);

  for (int l = 0; l < 4; ++l) {
    // forward rDFT over T:  ARI[4096 x 64] = H[4096 x 128] * BF[128 x 64]
    gemm(H, 0, 128, 0, BF, 0, 64, 0, ARI, 0, 64, 4096, 64, 128, 1,
         nullptr, nullptr, nullptr, nullptr, 1.f, nullptr, 0, 0, 0, 0);
    fft16n<<<256, 512, 0, stream>>>(ARI, XF, -1.f, 1.f);
    modemix<<<512, 256, 0, stream>>>(XF, swr1, swi1, swr2, swi2, l, OF);
    fft16n<<<256, 512, 0, stream>>>(OF, FRI, 1.f, 1.0f/16.0f);
    // inverse rDFT over T: SPEC[4096 x 128] = FRI[4096 x 64] * BI[64 x 128]
    gemm(FRI, 0, 64, 0, BI, 0, 128, 0, SPEC, 0, 128, 4096, 128, 64, 1,
         nullptr, nullptr, nullptr, nullptr, 1.f, nullptr, 0, 0, 0, 0);
    // H2 = gelu(skip_w*H + skip_b + SPEC)
    gemm(skip_w + l*4096, 0, 64, 0, H, Hb, 2048, 0, H2, Hb, 2048, 64, 2048, 64, 4,
         skip_b + l*64, nullptr, nullptr, nullptr, 1.f, SPEC, Hb, 2048, 0, 1);
    // MB = gelu(mlp_w1*H2 + b1)
    gemm(mlp_w1 + l*4096, 0, 64, 0, H2, Hb, 2048, 0, MB, Hb, 2048, 64, 2048, 64, 4,
         mlp_b1 + l*64, nullptr, nullptr, nullptr, 1.f, nullptr, 0, 0, 0, 1);
    // H = mlp_w2*MB + b2 + H2
    gemm(mlp_w2 + l*4096, 0, 64, 0, MB, Hb, 2048, 0, H, Hb, 2048, 64, 2048, 64, 4,
         mlp_b2 + l*64, nullptr, nullptr, nullptr, 1.f, H2, Hb, 2048, 0, 0);
  }

  // Conv1d + BN + GELU -> NS [B*N, 119, 64]
  conv_loc<<<dim3(8, 4, 64), 32, 0, stream>>>(H, conv_w, conv_b, bn_g, bn_b, NS);

  // q/k projections (bias per output feature = per column)
  gemm(NS, 0, 64, 0, wq, 0, 64, 1, QB, 0, 64, 7616, 64, 64, 1,
       nullptr, bq, nullptr, nullptr, 1.f, nullptr, 0, 0, 0, 0);
  gemm(NS, 0, 64, 0, wk, 0, 64, 1, KB, 0, 64, 7616, 64, 64, 1,
       nullptr, bk, nullptr, nullptr, 1.f, nullptr, 0, 0, 0, 0);

  // logits[b] = Q_b[16 x 7616] * K_b^T
  gemm(QB, 121856, 7616, 0, KB, 121856, 7616, 1, LG, 256, 16, 16, 16, 7616, 4,
       nullptr, nullptr, nullptr, nullptr, 1.f, nullptr, 0, 0, 0, 0);
  cl_kernel<<<4, 256, 0, stream>>>(LG, ln_g, ln_b, lsc, cbias, CLW, out);

  // qkv = st * in_w^T + in_b   (st = H^T per batch)
  gemm(H, Hb, 2048, 1, in_w, 0, 64, 1, QKV, 393216, 192, 2048, 192, 64, 4,
       nullptr, in_b, nullptr, nullptr, 1.f, nullptr, 0, 0, 0, 0);

  attn_flash<<<dim3(128, 4, 4), 32, 0, stream>>>(QKV, CLW, AT);

  // output projection -> MB [B*2048, 64]
  gemm(AT, 0, 64, 0, out_w, 0, 64, 1, MB, 0, 64, 8192, 64, 64, 1,
       nullptr, out_b, nullptr, nullptr, 1.f, nullptr, 0, 0, 0, 0);

  // head MLP: h1 over comb=[H ; ctx] split into two accumulating GEMMs
  gemm(h1_w, 0, 128, 0, H, Hb, 2048, 0, Z1, 262144, 2048, 128, 2048, 64, 4,
       nullptr, nullptr, nullptr, nullptr, 1.f, nullptr, 0, 0, 0, 0);
  gemm(h1_w + 64, 0, 128, 0, MB, Hb, 64, 1, Z1, 262144, 2048, 128, 2048, 64, 4,
       h1_b, nullptr, hbn1_g, hbn1_b, bnmul, nullptr, 0, 0, 1, 1);
  gemm(h2_w, 0, 128, 0, Z1, 262144, 2048, 0, Z2, Hb, 2048, 64, 2048, 128, 4,
       h2_b, nullptr, hbn2_g, hbn2_b, bnmul, nullptr, 0, 0, 0, 1);
  gemm(h3_w, 0, 64, 0, Z2, Hb, 2048, 0, Z3, 4096, 2048, 2, 2048, 64, 4,
       h3_b, nullptr, nullptr, nullptr, 1.f, nullptr, 0, 0, 0, 0);

  split_out<<<64, 256, 0, stream>>>(Z3, out);
}